// RejectionSampler_33895881900562
// MI455X (gfx1250) — compile-verified
//
#include <hip/hip_runtime.h>

// Rejection sampler (greedy spec-decode verification) for MI455X / gfx1250.
// Phase 1: row-wise argmax over V=128000 (memory-bound, ~590 MB @ 23.3 TB/s ~ 25us)
//          b128 non-temporal loads + wave32 shuffle reduction.
//          Unroll factor 5 divides the 125-iteration trip count exactly:
//          no remainder bookkeeping, 5-load s_clause per iteration.
// Phase 2: per-row prefix accept logic (trivial).

namespace {
constexpr int kB = 128;
constexpr int kK = 8;
constexpr int kV = 128000;
constexpr int kRows = kB * (kK + 1);   // 1152
constexpr int kInvalid = -1;
constexpr int kThreads = 256;          // 8 wave32s per row
constexpr int kVec = kV / 4;           // 32000 float4 per row; 32000/256 = 125 = 25*5
constexpr int kWaves = kThreads / 32;
}  // namespace

typedef float v4f __attribute__((ext_vector_type(4)));

__global__ __launch_bounds__(kThreads) void rowwise_argmax_kernel(
    const float* __restrict__ probs, int* __restrict__ row_argmax) {
  const int row = blockIdx.x;
  const v4f* __restrict__ rowp =
      reinterpret_cast<const v4f*>(probs + (size_t)row * kV);
  const int tid = threadIdx.x;

  float best_v = -__builtin_inff();
  int best_i = 0x7fffffff;

  // Each thread scans a strictly increasing index subsequence; strict '>'
  // keeps the first occurrence of the max within the thread.
#pragma unroll 5
  for (int i = tid; i < kVec; i += kThreads) {
    v4f v = __builtin_nontemporal_load(rowp + i);  // single-use stream: NT hint
    const int base = i << 2;
    if (v.x > best_v) { best_v = v.x; best_i = base; }
    if (v.y > best_v) { best_v = v.y; best_i = base + 1; }
    if (v.z > best_v) { best_v = v.z; best_i = base + 2; }
    if (v.w > best_v) { best_v = v.w; best_i = base + 3; }
  }

  // wave32 butterfly reduction; tie-break: smaller index wins (jnp.argmax).
#pragma unroll
  for (int off = 16; off > 0; off >>= 1) {
    float ov = __shfl_xor(best_v, off, 32);
    int oi = __shfl_xor(best_i, off, 32);
    if (ov > best_v || (ov == best_v && oi < best_i)) { best_v = ov; best_i = oi; }
  }

  __shared__ float s_v[kWaves];
  __shared__ int s_i[kWaves];
  const int wave = tid >> 5;
  const int lane = tid & 31;
  if (lane == 0) { s_v[wave] = best_v; s_i[wave] = best_i; }
  __syncthreads();

  if (wave == 0) {
    best_v = (lane < kWaves) ? s_v[lane] : -__builtin_inff();
    best_i = (lane < kWaves) ? s_i[lane] : 0x7fffffff;
#pragma unroll
    for (int off = 4; off > 0; off >>= 1) {
      float ov = __shfl_xor(best_v, off, 32);
      int oi = __shfl_xor(best_i, off, 32);
      if (ov > best_v || (ov == best_v && oi < best_i)) { best_v = ov; best_i = oi; }
    }
    if (lane == 0) row_argmax[row] = best_i;
  }
}

// Per-row accept logic. With t = target greedy tokens (always >= 0, so the
// reference's valid_mask is identically true), the reference reduces to:
//   r = number of leading positions with t[i] == draft[i]  (0..K)
//   out[i] = t[i] for i <= r, else -1
__global__ __launch_bounds__(kB) void accept_kernel(
    const int* __restrict__ row_argmax,  // [B, K+1]
    const int* __restrict__ draft,       // [B, K]
    int* __restrict__ out) {             // [B, K+1]
  const int b = threadIdx.x;
  if (b >= kB) return;
  const int* t = row_argmax + b * (kK + 1);
  const int* d = draft + b * kK;
  int* o = out + b * (kK + 1);
  int r = 0;
  while (r < kK && t[r] == d[r]) ++r;
#pragma unroll
  for (int i = 0; i <= kK; ++i) o[i] = (i <= r) ? t[i] : kInvalid;
}

extern "C" void kernel_launch(void* const* d_in, const int* in_sizes, int n_in,
                              void* d_out, int out_size, void* d_ws, size_t ws_size,
                              hipStream_t stream) {
  const int* draft_token_ids = (const int*)d_in[0];   // [128, 8] int32
  const float* target_probs = (const float*)d_in[1];  // [1152, 128000] fp32
  int* out = (int*)d_out;                             // [128, 9] int32
  int* row_argmax = (int*)d_ws;                       // 1152 ints of scratch

  rowwise_argmax_kernel<<<kRows, kThreads, 0, stream>>>(target_probs, row_argmax);
  accept_kernel<<<1, kB, 0, stream>>>(row_argmax, draft_token_ids, out);
}